// ElasticBand_89910845375147
// MI455X (gfx1250) — compile-verified
//
#include <hip/hip_runtime.h>
#include <math.h>

typedef float __attribute__((ext_vector_type(4))) fvec4;   // native clang vector:
// lowers to single b128 loads/stores and is accepted by the nontemporal builtins.

// Exact pointee type expected by __builtin_amdgcn_global_load_async_to_lds_b128
// (from the compiler diagnostic): generic pointer to gcc-vector int4.
typedef int v4i __attribute__((vector_size(4 * sizeof(int))));

#define N_IMG   32
#define N_ATOM  400000
#define ROWF    (N_ATOM * 3)        // 1,200,000 floats per image row
#define ROWV4   (ROWF / 4)          // 300,000 fvec4 per image row
#define RED_TPB   256
#define RED_ITER  16
#define RED_CHUNK (RED_TPB * RED_ITER)                      // 4096 fvec4 per block
#define RED_BLOCKS ((ROWV4 + RED_CHUNK - 1) / RED_CHUNK)    // 74

#define K_MAX_C 0.1f
#define DLT_K_C 0.02f
#define EPS_C   0.1f
#define PI_C    3.14159265358979323846f

// ---- gfx1250 async global->LDS path (probe via __has_builtin; fallback is
// ---- the plain VGPR-load loop, so this file compiles either way). ---------
#if defined(__HIP_DEVICE_COMPILE__) && __has_builtin(__builtin_amdgcn_global_load_async_to_lds_b128)
#define USE_ASYNC 1
#else
#define USE_ASYNC 0
#endif

#if defined(__HIP_DEVICE_COMPILE__) && __has_builtin(__builtin_amdgcn_s_wait_asynccnt)
#define WAIT_ASYNC(n) __builtin_amdgcn_s_wait_asynccnt(n)
#else
#define WAIT_ASYNC(n) asm volatile("s_wait_asynccnt %0" :: "i"(n) : "memory")
#endif

// ---------------------------------------------------------------------------
// Pass 0: zero the 30x6 double accumulators (must happen every call: pass 1
// accumulates with atomics, and the harness does not re-poison d_ws).
// ---------------------------------------------------------------------------
__global__ void neb_init(double* __restrict__ sums) {
    int i = threadIdx.x;
    if (i < 30 * 6) sums[i] = 0.0;
}

// ---------------------------------------------------------------------------
// Pass 1: per interior image m (0..29), accumulate the 6 primitive reductions
//   Tpp, Tmm, Tpm, Fp, Fm, Fff  over the 1.2M-float row.
// Full-chunk blocks stream the four b128 read streams through the CDNA5 async
// global->LDS engine (double-buffered, per-lane-private LDS slots, so a
// per-wave s_wait_asynccnt is the only synchronization needed). The ragged
// tail block uses plain bounds-checked loads.
// f32 thread partials -> f32 block tree -> f64 global atomic combine.
// ---------------------------------------------------------------------------
__global__ __launch_bounds__(RED_TPB)
void neb_reduce(const float* __restrict__ pos, const float* __restrict__ frc,
                double* __restrict__ sums) {
    const int m   = blockIdx.y;       // interior index 0..29
    const int img = m + 1;
    const fvec4* p0 = (const fvec4*)(pos + (size_t)(img - 1) * ROWF);
    const fvec4* p1 = (const fvec4*)(pos + (size_t)(img    ) * ROWF);
    const fvec4* p2 = (const fvec4*)(pos + (size_t)(img + 1) * ROWF);
    const fvec4* f1 = (const fvec4*)(frc + (size_t)(img    ) * ROWF);

    const int tid  = threadIdx.x;
    const int base = blockIdx.x * RED_CHUNK + tid;

    float tpp = 0.f, tmm = 0.f, tpm = 0.f, sfp = 0.f, sfm = 0.f, sff = 0.f;

#if USE_ASYNC
    // [buf][stream][lane]: 2 * 4 * 256 * 16B = 32 KB (320 KB available/WGP)
    __shared__ fvec4 stage[2][4][RED_TPB];

    if (blockIdx.x < RED_BLOCKS - 1) {
        // ---- async double-buffered fast path (uniform full chunk) ----
        auto issue = [&](int it) {
            const int j   = base + it * RED_TPB;
            const int buf = it & 1;
            __builtin_amdgcn_global_load_async_to_lds_b128(
                (v4i*)&p0[j], (v4i*)&stage[buf][0][tid], 0, 0);
            __builtin_amdgcn_global_load_async_to_lds_b128(
                (v4i*)&p1[j], (v4i*)&stage[buf][1][tid], 0, 0);
            __builtin_amdgcn_global_load_async_to_lds_b128(
                (v4i*)&p2[j], (v4i*)&stage[buf][2][tid], 0, 0);
            __builtin_amdgcn_global_load_async_to_lds_b128(
                (v4i*)&f1[j], (v4i*)&stage[buf][3][tid], 0, 0);
        };
        issue(0);
        issue(1);
        #pragma unroll
        for (int it = 0; it < RED_ITER; ++it) {
            const int buf = it & 1;
            // async loads complete in order: <=4 outstanding means the older
            // buffer's 4 transfers have landed in LDS.
            if (it + 1 < RED_ITER) WAIT_ASYNC(4);
            else                   WAIT_ASYNC(0);
            const fvec4 a = stage[buf][0][tid];
            const fvec4 b = stage[buf][1][tid];
            const fvec4 c = stage[buf][2][tid];
            const fvec4 f = stage[buf][3][tid];
            const fvec4 t = b - a;            // tau_p components
            const fvec4 u = c - b;            // tau_m components
            tpp += t.x * t.x + t.y * t.y + t.z * t.z + t.w * t.w;
            tmm += u.x * u.x + u.y * u.y + u.z * u.z + u.w * u.w;
            tpm += t.x * u.x + t.y * u.y + t.z * u.z + t.w * u.w;
            sfp += f.x * t.x + f.y * t.y + f.z * t.z + f.w * t.w;
            sfm += f.x * u.x + f.y * u.y + f.z * u.z + f.w * u.w;
            sff += f.x * f.x + f.y * f.y + f.z * f.z + f.w * f.w;
            if (it + 2 < RED_ITER) {
                // drain this wave's ds_loads before refilling the same buffer
                asm volatile("s_wait_dscnt 0x0" ::: "memory");
                issue(it + 2);
            }
        }
    } else
#endif
    {
        // ---- ragged tail (or no-async toolchain): plain loads ----
        #pragma unroll
        for (int it = 0; it < RED_ITER; ++it) {
            const int j = base + it * RED_TPB;
            if ((it & 3) == 0) {
                // gfx1250 prefetch path (global_prefetch_b8): warm next chunk.
                const int jp = j + RED_CHUNK;
                if (jp < ROWV4) {
                    __builtin_prefetch(&p0[jp], 0, 3);
                    __builtin_prefetch(&p1[jp], 0, 3);
                    __builtin_prefetch(&p2[jp], 0, 3);
                    __builtin_prefetch(&f1[jp], 0, 3);
                }
            }
            if (j < ROWV4) {
                const fvec4 a = p0[j], b = p1[j], c = p2[j], f = f1[j];
                const fvec4 t = b - a;
                const fvec4 u = c - b;
                tpp += t.x * t.x + t.y * t.y + t.z * t.z + t.w * t.w;
                tmm += u.x * u.x + u.y * u.y + u.z * u.z + u.w * u.w;
                tpm += t.x * u.x + t.y * u.y + t.z * u.z + t.w * u.w;
                sfp += f.x * t.x + f.y * t.y + f.z * t.z + f.w * t.w;
                sfm += f.x * u.x + f.y * u.y + f.z * u.z + f.w * u.w;
                sff += f.x * f.x + f.y * f.y + f.z * f.z + f.w * f.w;
            }
        }
    }

    __shared__ float sh[6][RED_TPB];
    sh[0][tid] = tpp; sh[1][tid] = tmm; sh[2][tid] = tpm;
    sh[3][tid] = sfp; sh[4][tid] = sfm; sh[5][tid] = sff;
    __syncthreads();
    for (int off = RED_TPB / 2; off > 0; off >>= 1) {
        if (tid < off) {
            #pragma unroll
            for (int q = 0; q < 6; ++q) sh[q][tid] += sh[q][tid + off];
        }
        __syncthreads();
    }
    if (tid == 0) {
        #pragma unroll
        for (int q = 0; q < 6; ++q)
            atomicAdd(&sums[m * 6 + q], (double)sh[q][0]);   // global_atomic_add_f64
    }
}

// ---------------------------------------------------------------------------
// Pass 2: one wave; turn eng[32] + sums[30][6] into per-image (A, B, C).
//   frc_neb = A*tau_p + B*tau_m + C*frc
// ---------------------------------------------------------------------------
__global__ void neb_scalars(const float* __restrict__ eng,
                            const double* __restrict__ sums,
                            float* __restrict__ coef) {
    __shared__ float e[N_IMG];
    __shared__ float kk[N_IMG - 1];
    __shared__ float semax;
    __shared__ float seref;
    __shared__ int   simax;

    const int t = threadIdx.x;
    if (t < N_IMG) e[t] = eng[t];
    __syncthreads();

    if (t == 0) {
        float mn = e[0], mx = e[0];
        int am = 0;
        for (int j = 1; j < N_IMG; ++j) {
            mn = fminf(mn, e[j]);
            if (e[j] > mx) { mx = e[j]; am = j; }   // first argmax, like jnp.argmax
        }
        semax = mx;
        seref = mn - EPS_C;
        simax = am;
    }
    __syncthreads();

    const float emax = semax, eref = seref;
    const int   imax = simax;

    if (t < N_IMG - 1) {
        const float ei = fmaxf(e[t + 1], e[t]);
        float kv = K_MAX_C - DLT_K_C * (emax - ei) / (emax - eref);
        if (ei < eref) kv = K_MAX_C - DLT_K_C;
        kk[t] = kv;
    }
    __syncthreads();

    if (t < N_IMG - 2) {
        const int m = t;
        const float e0 = e[m], e1 = e[m + 1], e2 = e[m + 2];
        const float k0 = kk[m], k1 = kk[m + 1];

        const float mpp = (e2 > e1 && e1 > e0) ? 1.f : 0.f;
        const float mmm = (e2 < e1 && e1 < e0) ? 1.f : 0.f;
        const float nb  = 1.f - fmaxf(mpp, mmm);
        const float mp  = ((e2 > e1) ? 1.f : 0.f) * nb;
        const float mm  = ((e2 < e1) ? 1.f : 0.f) * nb;
        const float d21 = fabsf(e2 - e1), d01 = fabsf(e0 - e1);
        const float dvmax = fmaxf(d21, d01), dvmin = fminf(d21, d01);
        // tau = a*tau_p + b*tau_m
        const float a = mpp + dvmax * mp + dvmin * mm;
        const float b = mmm + dvmin * mp + dvmax * mm;

        const float Tpp = (float)sums[m * 6 + 0];
        const float Tmm = (float)sums[m * 6 + 1];
        const float Tpm = (float)sums[m * 6 + 2];
        const float Fp  = (float)sums[m * 6 + 3];
        const float Fm  = (float)sums[m * 6 + 4];
        const float Fff = (float)sums[m * 6 + 5];

        const float nt   = sqrtf(a * a * Tpp + 2.f * a * b * Tpm + b * b * Tmm); // ||tau||
        const float p    = (a * Fp + b * Fm) / nt;                  // F . tau_hat
        const float s    = k1 * sqrtf(Tmm) - k0 * sqrtf(Tpp);       // spr_para scale
        const float Stau = (k1 * (a * Tpm + b * Tmm) - k0 * (a * Tpp + b * Tpm)) / nt;
        const float SF   = k1 * Fm - k0 * Fp;                       // spr . frc
        const float Fperp2 = Fff - p * p;
        const float SS     = k1 * k1 * Tmm - 2.f * k0 * k1 * Tpm + k0 * k0 * Tpp;
        const float Sperp2 = SS - 2.f * s * Stau + s * s;
        const float d  = SF - p * Stau;                             // <S_perp, F_perp>
        const float sw = (2.f / PI_C) * atan2f(Fperp2, Sperp2);
        const float g  = d * sw;
        // frc_para doubling: reference indexes the length-30 array with the
        // GLOBAL argmax index (replicated verbatim; OOB updates are dropped).
        const float cp = (m == imax) ? 2.f : 1.f;
        const float fac = (g - cp) * p / nt;

        coef[m * 4 + 0] = -k0 + fac * a;   // A on tau_p
        coef[m * 4 + 1] =  k1 + fac * b;   // B on tau_m
        coef[m * 4 + 2] = -g;              // C on frc
        coef[m * 4 + 3] = 0.f;
    }
}

// ---------------------------------------------------------------------------
// Pass 3: out = frc + A*(pos[i]-pos[i-1]) + B*(pos[i+1]-pos[i]) + C*frc
// (out = frc at the two end images). Non-temporal stores: output is
// write-once, keep it out of the way of the pos/frc read streams.
// ---------------------------------------------------------------------------
__global__ __launch_bounds__(RED_TPB)
void neb_apply(const float* __restrict__ pos, const float* __restrict__ frc,
               const float* __restrict__ coef, float* __restrict__ out) {
    const int img  = blockIdx.y;
    const int base = blockIdx.x * RED_CHUNK + threadIdx.x;
    const fvec4* f1 = (const fvec4*)(frc + (size_t)img * ROWF);
    fvec4*       o  = (fvec4*)(out + (size_t)img * ROWF);

    if (img == 0 || img == N_IMG - 1) {
        #pragma unroll
        for (int it = 0; it < RED_ITER; ++it) {
            const int j = base + it * RED_TPB;
            if (j < ROWV4) {
                fvec4 f = f1[j];
                __builtin_nontemporal_store(f, &o[j]);
            }
        }
        return;
    }

    const int m = img - 1;
    const float A = coef[m * 4 + 0];
    const float B = coef[m * 4 + 1];
    const float C = coef[m * 4 + 2];
    const fvec4* p0 = (const fvec4*)(pos + (size_t)(img - 1) * ROWF);
    const fvec4* p1 = (const fvec4*)(pos + (size_t)(img    ) * ROWF);
    const fvec4* p2 = (const fvec4*)(pos + (size_t)(img + 1) * ROWF);

    #pragma unroll
    for (int it = 0; it < RED_ITER; ++it) {
        const int j = base + it * RED_TPB;
        if ((it & 3) == 0) {
            const int jp = j + RED_CHUNK;
            if (jp < ROWV4) {
                __builtin_prefetch(&p0[jp], 0, 3);
                __builtin_prefetch(&p1[jp], 0, 3);
                __builtin_prefetch(&p2[jp], 0, 3);
                __builtin_prefetch(&f1[jp], 0, 3);
            }
        }
        if (j < ROWV4) {
            const fvec4 a = p0[j], b = p1[j], c = p2[j], f = f1[j];
            const fvec4 r = f + A * (b - a) + B * (c - b) + C * f;
            __builtin_nontemporal_store(r, &o[j]);
        }
    }
}

// ---------------------------------------------------------------------------
extern "C" void kernel_launch(void* const* d_in, const int* in_sizes, int n_in,
                              void* d_out, int out_size, void* d_ws, size_t ws_size,
                              hipStream_t stream) {
    (void)in_sizes; (void)n_in; (void)out_size; (void)ws_size;
    const float* pos = (const float*)d_in[0];
    const float* frc = (const float*)d_in[1];
    const float* eng = (const float*)d_in[2];
    float*       out = (float*)d_out;

    double* sums = (double*)d_ws;                       // 30*6 doubles = 1440 B
    float*  coef = (float*)((char*)d_ws + 1536);        // 30*4 floats

    neb_init   <<<1, 256, 0, stream>>>(sums);
    neb_reduce <<<dim3(RED_BLOCKS, 30), RED_TPB, 0, stream>>>(pos, frc, sums);
    neb_scalars<<<1, 32, 0, stream>>>(eng, sums, coef);
    neb_apply  <<<dim3(RED_BLOCKS, N_IMG), RED_TPB, 0, stream>>>(pos, frc, coef, out);
}